// RoutingByAgreement_77833397338704
// MI455X (gfx1250) — compile-verified
//
#include <hip/hip_runtime.h>
#include <hip/hip_bf16.h>
#include <math.h>

typedef __attribute__((ext_vector_type(2))) float v2f;
typedef __attribute__((ext_vector_type(8))) float v8f;
typedef __attribute__((ext_vector_type(4))) unsigned int v4u;
typedef __attribute__((ext_vector_type(8))) int v8i;
typedef __attribute__((ext_vector_type(4))) int v4i;

#define NCAPS 64
#define DDIM 64
#define ROWSTRIDE 66                      // 64 data dwords + 2 pad dwords (TDM pad) per row
#define TILE_DWORDS (NCAPS * ROWSTRIDE)   // 4224 dwords = 16896 B per batch tile
#define WAVES_PER_BLOCK 4
#define EPS 1e-8f

__global__ __launch_bounds__(32 * WAVES_PER_BLOCK, 1)
void routing_by_agreement_kernel(const float* __restrict__ caps,
                                 float* __restrict__ out,
                                 int nbatch) {
    __shared__ float tiles[WAVES_PER_BLOCK][TILE_DWORDS];
    __shared__ float bvec[WAVES_PER_BLOCK][NCAPS];
    __shared__ float cvec[WAVES_PER_BLOCK][NCAPS];
    __shared__ float vvec[WAVES_PER_BLOCK][DDIM];

    const int lane = threadIdx.x & 31;
    const int wave = __builtin_amdgcn_readfirstlane((int)(threadIdx.x >> 5));
    const int batch = blockIdx.x * WAVES_PER_BLOCK + wave;
    if (batch >= nbatch) return;   // wave-uniform branch (skips TDM issue entirely)

    float* T  = &tiles[wave][0];
    float* bv = &bvec[wave][0];
    float* cv = &cvec[wave][0];
    float* vv = &vvec[wave][0];

    // ---------------- TDM: DMA one 64x64 f32 tile into LDS, 2-dword pad per row ----
    {
        unsigned long long gaddr =
            (unsigned long long)(const void*)(caps + (size_t)batch * (NCAPS * DDIM));
        unsigned int lds_byte = (unsigned int)(size_t)(void*)T;  // low 32 bits = LDS offset

        v4u g0;
        g0.x = 1u;                                        // count=1 (valid user descriptor)
        g0.y = lds_byte;                                  // lds_addr
        g0.z = (unsigned int)(gaddr & 0xFFFFFFFFull);     // global_addr[31:0]
        g0.w = (unsigned int)((gaddr >> 32) & 0x1FFFFFFull) | (2u << 30); // addr[56:32] | type=2

        v8i g1;
        // workgroup_mask=0 | data_size=2 (4B) | pad_enable | pad_interval=5 (64 dw) | pad_amount=1 (2 dw)
        g1[0] = (int)((2u << 16) | (1u << 20) | (5u << 22) | (1u << 25));
        g1[1] = (int)(((unsigned)(NCAPS * DDIM) & 0xFFFFu) << 16); // tensor_dim0[15:0]=4096
        g1[2] = (int)(1u << 16);                                   // tensor_dim0 hi=0, tensor_dim1=1
        g1[3] = (int)(((unsigned)(NCAPS * DDIM)) << 16);           // tile_dim0=4096 (1D tile)
        g1[4] = 1;                                                 // tile_dim1=1, tile_dim2=0
        g1[5] = NCAPS * DDIM;                                      // tensor_dim0_stride=4096
        g1[6] = 0;
        g1[7] = 0;
        v4i z4 = {0, 0, 0, 0};
        v8i z8 = {0, 0, 0, 0, 0, 0, 0, 0};

        // 6-arg form (clang-23 / amdgpu-toolchain): (g0, g1, g2, g3, ext, cpol)
        __builtin_amdgcn_tensor_load_to_lds(g0, g1, z4, z4, z8, 0);
        __builtin_amdgcn_s_wait_tensorcnt(0);
        asm volatile("" ::: "memory");
    }

    // logits b = 0
    bv[lane]      = 0.0f;
    bv[lane + 32] = 0.0f;

    const int g = lane >> 4;        // half-wave group for WMMA fragments
    const int m = lane & 15;        // row within 16-row block

    #pragma unroll
    for (int it = 0; it < 3; ++it) {
        // -------- softmax(b) over N=64 (redundant per lane via LDS broadcast reads) --
        float mx = -INFINITY;
        #pragma unroll 8
        for (int n = 0; n < NCAPS; ++n) mx = fmaxf(mx, bv[n]);
        float sum = 0.0f;
        #pragma unroll 8
        for (int n = 0; n < NCAPS; ++n) sum += __expf(bv[n] - mx);
        float inv = 1.0f / sum;
        cv[lane]      = __expf(bv[lane]      - mx) * inv;
        cv[lane + 32] = __expf(bv[lane + 32] - mx) * inv;

        // -------- s[d] = sum_n c[n] * M[n,d]; lane owns d = 2*lane, 2*lane+1 --------
        float s0 = 0.0f, s1 = 0.0f;
        #pragma unroll 8
        for (int n = 0; n < NCAPS; ++n) {
            float c  = cv[n];                                       // broadcast read
            float2 t = *(const float2*)(T + n * ROWSTRIDE + 2 * lane); // conflict-free b64
            s0 = fmaf(c, t.x, s0);
            s1 = fmaf(c, t.y, s1);
        }

        // -------- squash -------------------------------------------------------------
        float p = fmaf(s0, s0, s1 * s1);
        #pragma unroll
        for (int off = 16; off >= 1; off >>= 1) p += __shfl_xor(p, off, 32);
        float nrm   = sqrtf(p);
        float scale = p / (1.0f + p) / (nrm + EPS);
        float v0 = scale * s0, v1 = scale * s1;

        if (it == 2) {
            float2 o; o.x = v0; o.y = v1;
            *(float2*)(out + (size_t)batch * DDIM + 2 * lane) = o;
        } else {
            float2 o; o.x = v0; o.y = v1;
            *(float2*)(vv + 2 * lane) = o;

            // ---- agreement[n] = M[n,:] . v via V_WMMA_F32_16X16X4_F32 --------------
            // A: 16x4 chunk of M rows (lane layout: A[m][2g+p]); B: v broadcast over
            // all 16 columns (B[k][*] = v[k0+k], lane layout B[p]=v[k0+2g+p]).
            // D columns are identical; lanes m==0 (lanes 0,16) read agreement rows.
            #pragma unroll
            for (int rb = 0; rb < NCAPS; rb += 16) {
                v8f acc = {0.f, 0.f, 0.f, 0.f, 0.f, 0.f, 0.f, 0.f};
                #pragma unroll
                for (int k0 = 0; k0 < DDIM; k0 += 4) {
                    v2f a = *(const v2f*)(T + (rb + m) * ROWSTRIDE + k0 + 2 * g);
                    v2f b = *(const v2f*)(vv + k0 + 2 * g);
                    acc = __builtin_amdgcn_wmma_f32_16x16x4_f32(
                        false, a, false, b, (short)0, acc, false, false);
                }
                if (m == 0) {
                    #pragma unroll
                    for (int i = 0; i < 8; ++i)
                        bv[rb + 8 * g + i] += acc[i];
                }
            }
        }
    }
}

extern "C" void kernel_launch(void* const* d_in, const int* in_sizes, int n_in,
                              void* d_out, int out_size, void* d_ws, size_t ws_size,
                              hipStream_t stream) {
    const float* caps = (const float*)d_in[0];
    float* out = (float*)d_out;
    const int nbatch = in_sizes[0] / (NCAPS * DDIM);   // 16384
    const int blocks = (nbatch + WAVES_PER_BLOCK - 1) / WAVES_PER_BLOCK;
    hipLaunchKernelGGL(routing_by_agreement_kernel,
                       dim3(blocks), dim3(32 * WAVES_PER_BLOCK), 0, stream,
                       caps, out, nbatch);
}